// MFVIJointConstAndOneStageLinearCRFSRL_33904471834785
// MI455X (gfx1250) — compile-verified
//
#include <hip/hip_runtime.h>
#include <stdint.h>
#include <math.h>

#define SB 4
#define SS 128
#define SS2 (SS*SS)

typedef __attribute__((ext_vector_type(2))) float v2f;
typedef __attribute__((ext_vector_type(8))) float v8f;

__device__ __forceinline__ float2 ld2(const float* p) {
  return *reinterpret_cast<const float2*>(p);
}

// validf[b,i] = edge_mask[b,i,i] ? 1.0 : 0.0
__global__ void k_valid(const uint8_t* __restrict__ edge_mask,
                        float* __restrict__ validf) {
  int t = blockIdx.x * blockDim.x + threadIdx.x;
  if (t >= SB * SS) return;
  int b = t >> 7, i = t & (SS - 1);
  validf[t] = edge_mask[(size_t)b * SS2 + (size_t)i * SS + i] ? 1.f : 0.f;
}

// splitsum[b,i,s] = sum_j s_split[b,i,j,s] * span2o[b,i,j,s]  (precomputed once)
__global__ void k_splitsum(const float* __restrict__ s_split,
                           const float* __restrict__ validf,
                           float* __restrict__ splitsum) {
  int blk = blockIdx.x;
  int b = blk >> 7, i = blk & (SS - 1);
  int s = threadIdx.x;
  const float* vf = validf + b * SS;
  float v0 = vf[0];
  int ip = (i - 1) & (SS - 1);
  float vip = vf[ip];
  float vs = vf[s];
  float cm0s = v0 * vs * (s > 0 ? 1.f : 0.f);
  float sum = 0.f;
  const float* base = s_split + ((size_t)(b * SS + i)) * SS2 + s;
  for (int j = 0; j < SS; ++j) {
    float vj = vf[j];
    float chart_r = vip * vj * (ip < j ? 1.f : 0.f);
    float cm0j = v0 * vj * (j > 0 ? 1.f : 0.f);
    int mx = i > j ? i : j;
    float msk = chart_r * cm0j * cm0s * (mx != s ? 1.f : 0.f);
    sum += msk * base[(size_t)j * SS];
  }
  splitsum[(size_t)b * SS2 + i * SS + s] = sum;
}

// q_edge = s_edge ; q_span = triu(s_const) + triu(s_const,1)^T
__global__ void k_qinit(const float* __restrict__ s_edge,
                        const float* __restrict__ s_const,
                        float* __restrict__ q_edge, float* __restrict__ q_span) {
  int idx = blockIdx.x * blockDim.x + threadIdx.x;
  if (idx >= SB * SS2) return;
  q_edge[idx] = s_edge[idx];
  int b = idx / SS2, r = idx % SS2, i = r / SS, j = r % SS;
  float v = (i <= j) ? s_const[(size_t)b * SS2 + i * SS + j]
                     : s_const[(size_t)b * SS2 + j * SS + i];
  q_span[idx] = v;
}

__global__ void k_sigmoid(const float* __restrict__ q_edge,
                          const float* __restrict__ q_span,
                          float* __restrict__ pe, float* __restrict__ ps) {
  int idx = blockIdx.x * blockDim.x + threadIdx.x;
  if (idx >= SB * SS2) return;
  pe[idx] = 1.f / (1.f + expf(-q_edge[idx]));
  ps[idx] = 1.f / (1.f + expf(-q_span[idx]));
}

// One block per (b,row). Wave w owns output tile m0=16w. For each K=4 chunk of s,
// each lane builds its W[m,s]=tensor*mask*weight entries in the 16x4 f32 A layout
// (lanes 0-15: K=0,1 ; lanes 16-31: K=2,3), then V_WMMA_F32_16X16X4_F32 with an
// all-ones B accumulates row sums into the 16x16 f32 accumulator. All masking is
// branchless {0,1}-float arithmetic so EXEC stays all-ones and loads stay b64.
__global__ __launch_bounds__(256) void k_update(
    const float* __restrict__ sib, const float* __restrict__ cop,
    const float* __restrict__ grd, const float* __restrict__ ep,
    const float* __restrict__ hb,  const float* __restrict__ he,
    const float* __restrict__ pe,  const float* __restrict__ ps,
    const float* __restrict__ splitsum, const float* __restrict__ validf,
    float* __restrict__ q_edge, float* __restrict__ q_span) {
  int blk = blockIdx.x;
  int b = blk >> 7;
  int r = blk & (SS - 1);                 // h for edge terms, i for span terms
  int lane = threadIdx.x & 31;
  int wave = threadIdx.x >> 5;
  int m0 = wave * 16;
  int m = m0 + (lane & 15);               // m for edge terms, j for span terms
  int shalf = (lane >> 4) << 1;           // K offset: lanes 0-15 -> K0,1 ; 16-31 -> K2,3

  const float* vf = validf + b * SS;
  float vhf = vf[r];
  float vmf = vf[m];
  float vhmf = vhf * vmf;

  const float* peB = pe + (size_t)b * SS2;
  const float* psB = ps + (size_t)b * SS2;

  size_t te = ((size_t)(b * SS + r) * SS + m) * SS;   // [b,r,m,:]
  const float* sibp = sib + te;
  const float* copp = cop + te;
  const float* grdp = grd + te;
  const float* epp  = ep + te;
  const float* perow = peB + r * SS;                  // pe[b,r,:]
  const float* psrow = psB + m * SS;                  // ps[b,m,:]
  const float* vrow  = vf;                            // validf[b,:]

  v2f bones = {1.f, 1.f};
  v8f acc_e = {};

  for (int s0 = 0; s0 < SS; s0 += 4) {
    int sA = s0 + shalf;
    float2 sv = ld2(sibp + sA);
    float2 cv = ld2(copp + sA);
    float2 gv = ld2(grdp + sA);
    float2 ev = ld2(epp + sA);
    float2 pr = ld2(perow + sA);
    float2 qr = ld2(psrow + sA);
    float2 vv = ld2(vrow + sA);
    v2f a;
#pragma unroll
    for (int t = 0; t < 2; ++t) {
      int ss = sA + t;
      float vse = t ? vv.y : vv.x;
      float base = vhmf * vse * (m != ss ? 1.f : 0.f);
      float f_em = (r != ss) ? 1.f : 0.f;             // extra em2o factor
      int mn = m < ss ? m : ss, mx = m < ss ? ss : m;
      float f_nc = ((mn >= r) || (mx <= r)) ? 1.f : 0.f;  // noncross factor
      float sve = t ? sv.y : sv.x;
      float cve = t ? cv.y : cv.x;
      float gve = t ? gv.y : gv.x;
      float eve = t ? ev.y : ev.x;
      float pre = t ? pr.y : pr.x;
      float qre = t ? qr.y : qr.x;
      float pc = peB[(size_t)ss * SS + m];            // pe[b,s,m]
      a[t] = base * (f_em * (sve * pre + cve * pc + gve * qre) +
                     f_nc * (eve * qre));
    }
    acc_e = __builtin_amdgcn_wmma_f32_16x16x4_f32(
        false, a, false, bones, (short)0, acc_e, false, false);
  }

  // ---- span terms (i=r, j=m) ----
  int ip = (r - 1) & (SS - 1);
  float v0 = vf[0];
  float chart_r_ij = vf[ip] * vmf * (ip < m ? 1.f : 0.f);
  float cm0i = v0 * vhf * (r > 0 ? 1.f : 0.f);
  float cm0j = v0 * vmf * (m > 0 ? 1.f : 0.f);
  float pre_ij = chart_r_ij * cm0i * cm0j * (r != m ? 1.f : 0.f);
  int mxij = r > m ? r : m;

  const float* hbp = hb + te;                                   // hb[b,i,j,:]
  const float* hep = he + ((size_t)(b * SS + m) * SS + r) * SS; // he src [b,j,i,:]

  v8f acc_s = {};
  for (int s0 = 0; s0 < SS; s0 += 4) {
    int sA = s0 + shalf;
    float2 hv = ld2(hbp + sA);
    float2 tv = ld2(hep + sA);
    float2 vv = ld2(vrow + sA);
    v2f a;
#pragma unroll
    for (int t = 0; t < 2; ++t) {
      int ss = sA + t;
      float vse = t ? vv.y : vv.x;
      float cm0s = v0 * vse * (ss > 0 ? 1.f : 0.f);
      float f_in = ((r <= ss) && (m >= ss)) ? 0.f : 1.f;        // ~inside
      float outs = pre_ij * cm0s * (mxij != ss ? 1.f : 0.f) * f_in;
      float hve = t ? hv.y : hv.x;
      float tve = t ? tv.y : tv.x;
      float pci = peB[(size_t)ss * SS + r];           // pe[b,s,i]
      float pcj = peB[(size_t)ss * SS + m];           // pe[b,s,j]
      a[t] = outs * (hve * pci + tve * pcj);
    }
    acc_s = __builtin_amdgcn_wmma_f32_16x16x4_f32(
        false, a, false, bones, (short)0, acc_s, false, false);
  }

  // C/D layout: lane 0 holds rows M=0..7, lane 16 holds M=8..15 (all columns equal).
  if ((lane & 15) == 0) {
    int rbase = (lane >> 4) * 8;
    size_t qb = (size_t)b * SS2 + (size_t)r * SS;
#pragma unroll
    for (int rr = 0; rr < 8; ++rr) {
      int mm = m0 + rbase + rr;
      q_edge[qb + mm] += acc_e[rr];
      float extra = psB[r * SS + mm] * splitsum[(size_t)b * SS2 + r * SS + mm];
      q_span[qb + mm] += acc_s[rr] + extra;
    }
  }
}

extern "C" void kernel_launch(void* const* d_in, const int* in_sizes, int n_in,
                              void* d_out, int out_size, void* d_ws, size_t ws_size,
                              hipStream_t stream) {
  const float* s_edge  = (const float*)d_in[0];
  const float* s_const = (const float*)d_in[1];
  const float* s_sib   = (const float*)d_in[2];
  const float* s_cop   = (const float*)d_in[3];
  const float* s_grd   = (const float*)d_in[4];
  const float* s_ep    = (const float*)d_in[5];
  const float* s_split = (const float*)d_in[6];
  const float* s_hb    = (const float*)d_in[7];
  const float* s_he    = (const float*)d_in[8];
  const uint8_t* edge_mask = (const uint8_t*)d_in[9];

  float* q_edge = (float*)d_out;
  float* q_span = q_edge + (size_t)SB * SS2;

  char* ws = (char*)d_ws;
  float* pe = (float*)ws;        ws += (size_t)SB * SS2 * 4;
  float* ps = (float*)ws;        ws += (size_t)SB * SS2 * 4;
  float* splitsum = (float*)ws;  ws += (size_t)SB * SS2 * 4;
  float* validf = (float*)ws;

  k_valid<<<2, 256, 0, stream>>>(edge_mask, validf);
  k_splitsum<<<SB * SS, SS, 0, stream>>>(s_split, validf, splitsum);
  {
    int total = SB * SS2, thr = 256;
    k_qinit<<<(total + thr - 1) / thr, thr, 0, stream>>>(s_edge, s_const, q_edge, q_span);
  }
  for (int it = 0; it < 3; ++it) {
    int total = SB * SS2, thr = 256;
    k_sigmoid<<<(total + thr - 1) / thr, thr, 0, stream>>>(q_edge, q_span, pe, ps);
    k_update<<<SB * SS, 256, 0, stream>>>(s_sib, s_cop, s_grd, s_ep, s_hb, s_he,
                                          pe, ps, splitsum, validf, q_edge, q_span);
  }
}